// CompRAMBase_45629732552952
// MI455X (gfx1250) — compile-verified
//
#include <hip/hip_runtime.h>
#include <math.h>

// ---------------------------------------------------------------------------
// CompRAM relational graph-attention layer for MI455X (gfx1250, wave32).
// f32 WMMA (V_WMMA_F32_16X16X4_F32) for all 128x128 GEMMs, weights staged to
// LDS via GLOBAL_LOAD_ASYNC_TO_LDS_B128 (ASYNCcnt path) overlapped with the
// irregular row gathers; atomic segment softmax + scatter-add aggregation.
// ---------------------------------------------------------------------------

typedef float v2f __attribute__((ext_vector_type(2)));
typedef float v8f __attribute__((ext_vector_type(8)));

#define DIM     128
#define N_ENT   100000
#define N_EDGE  500000
#define N_REL2  474
#define BATCH   4096

// monotone float atomic max: int-max for non-negative, uint-min for negative.
__device__ __forceinline__ void atomic_max_float(float* addr, float val) {
  if (val >= 0.0f) atomicMax((int*)addr, __float_as_int(val));
  else             atomicMin((unsigned int*)addr, __float_as_uint(val));
}

// ---------------------------------------------------------------------------
// Async-stage the first 16 KB (rows 0-31) of a 128x128 f32 matrix into LDS.
// blockDim.x must be 256: each thread issues 4x b128 async copies (64 B).
// Caller must stage_w_wait() + __syncthreads() before reading LDS.
// ---------------------------------------------------------------------------
__device__ __forceinline__ void stage_w_issue(const float* __restrict__ src,
                                              float* lds) {
  const unsigned ldsBase = (unsigned)(size_t)lds;      // LDS aperture: low 32 bits = LDS addr
  const unsigned t = threadIdx.x * 16u;
#pragma unroll
  for (int i = 0; i < 4; ++i) {
    unsigned off = t + (unsigned)i * 4096u;
    asm volatile("global_load_async_to_lds_b128 %0, %1, %2"
                 :: "v"(ldsBase + off), "v"(off), "s"(src)
                 : "memory");
  }
}
__device__ __forceinline__ void stage_w_wait() {
  asm volatile("s_wait_asynccnt 0" ::: "memory");
}

// ---------------------------------------------------------------------------
// init: zero e_r/e_i accumulators (in d_out), segmax=-inf, denom=0
// ---------------------------------------------------------------------------
__global__ void init_kernel(float* __restrict__ eAcc, float* __restrict__ segmax,
                            float* __restrict__ denom) {
  size_t i = (size_t)blockIdx.x * blockDim.x + threadIdx.x;
  size_t stride = (size_t)gridDim.x * blockDim.x;
  const size_t nAcc = 2ull * N_ENT * DIM;
  for (size_t j = i; j < nAcc; j += stride) eAcc[j] = 0.0f;
  for (size_t j = i; j < N_ENT; j += stride) {
    segmax[j] = __int_as_float(0xff800000);  // -inf
    denom[j]  = 0.0f;
  }
}

// ---------------------------------------------------------------------------
// Generic C[M,128] = act(A[M,128] @ W[128,128]); W async-staged into LDS.
// One wave = one 16-row tile, 8 N-panels, K=128 via 32x wmma_f32_16x16x4_f32.
// ---------------------------------------------------------------------------
__global__ __launch_bounds__(256) void gemm128_kernel(
    const float* __restrict__ A, const float* __restrict__ Wm,
    float* __restrict__ C, int M, int act /*0=none,1=tanh*/) {
  __shared__ float lds[DIM * DIM];
  stage_w_issue(Wm, lds);                       // async: overlaps with A loads

  const int lane  = threadIdx.x & 31;
  const int wv    = threadIdx.x >> 5;
  const int mbase = (blockIdx.x * 8 + wv) * 16;
  const int koff  = (lane >> 4) << 1;           // lanes 16-31 carry K+2,K+3
  const int n0    = lane & 15;
  int m = mbase + n0; if (m >= M) m = M - 1;    // clamp rows; stores masked below

  v2f af[32];
  const float* ar = A + (size_t)m * DIM + koff;
#pragma unroll
  for (int ks = 0; ks < 32; ++ks) af[ks] = *(const v2f*)(ar + ks * 4);

  stage_w_wait();                               // our LDS writes complete
  __syncthreads();                              // everyone's writes visible
  if (mbase >= M) return;                       // wave-uniform exit (EXEC stays full)

  for (int p = 0; p < 8; ++p) {
    v8f acc = {0.f, 0.f, 0.f, 0.f, 0.f, 0.f, 0.f, 0.f};
    const float* wp = lds + (size_t)koff * DIM + p * 16 + n0;
#pragma unroll
    for (int ks = 0; ks < 32; ++ks) {
      v2f b; b.x = wp[ks * 512]; b.y = wp[ks * 512 + 128];
      acc = __builtin_amdgcn_wmma_f32_16x16x4_f32(
          false, af[ks], false, b, (short)0, acc, false, false);
    }
    const int mrow = mbase + ((lane >> 4) << 3);   // C: VGPR v -> M=v (+8 for hi half)
#pragma unroll
    for (int v = 0; v < 8; ++v) {
      int r = mrow + v;
      if (r < M) {
        float x = acc[v];
        C[(size_t)r * DIM + p * 16 + n0] = act ? tanhf(x) : x;
      }
    }
  }
}

// ---------------------------------------------------------------------------
// Per-edge: gather, complex compose, h_r = m_r@w_real, h_i = m_i@w_imag.
// Two async LDS phases (16 KB staged each); A-fragments live in VGPRs throughout.
// ---------------------------------------------------------------------------
__global__ __launch_bounds__(256) void edge_kernel(
    const int* __restrict__ edge_src, const int* __restrict__ edge_type,
    const float* __restrict__ entR, const float* __restrict__ entI,
    const float* __restrict__ relR, const float* __restrict__ relI,
    const float* __restrict__ wReal, const float* __restrict__ wImag,
    float* __restrict__ hR, float* __restrict__ hI) {
  __shared__ float lds[DIM * DIM];
  stage_w_issue(wReal, lds);                    // async: overlaps with gathers

  const int lane  = threadIdx.x & 31;
  const int wv    = threadIdx.x >> 5;
  const int mbase = (blockIdx.x * 8 + wv) * 16;
  const bool active = (mbase < N_EDGE);         // wave-uniform
  const int koff = (lane >> 4) << 1;
  const int n0   = lane & 15;

  // gather + complex composition (clamped indices; safe for inactive waves)
  int e = mbase + n0; if (e >= N_EDGE) e = N_EDGE - 1;
  const int s = edge_src[e];
  const int t = edge_type[e];
  const float* pr = entR + (size_t)s * DIM + koff;
  const float* pi = entI + (size_t)s * DIM + koff;
  const float* qr = relR + (size_t)t * DIM + koff;
  const float* qi = relI + (size_t)t * DIM + koff;
  v2f amr[32], ami[32];
#pragma unroll
  for (int ks = 0; ks < 32; ++ks) {
    v2f a = *(const v2f*)(pr + ks * 4);
    v2f b = *(const v2f*)(pi + ks * 4);
    v2f c = *(const v2f*)(qr + ks * 4);
    v2f d = *(const v2f*)(qi + ks * 4);
    amr[ks] = a * c - b * d;                    // complex composition
    ami[ks] = a * d + b * c;
  }

  // ---- phase 1: w_real -> h_r ----
  stage_w_wait();
  __syncthreads();
  if (active) {
    for (int p = 0; p < 8; ++p) {
      v8f acc = {0.f, 0.f, 0.f, 0.f, 0.f, 0.f, 0.f, 0.f};
      const float* wp = lds + (size_t)koff * DIM + p * 16 + n0;
#pragma unroll
      for (int ks = 0; ks < 32; ++ks) {
        v2f b; b.x = wp[ks * 512]; b.y = wp[ks * 512 + 128];
        acc = __builtin_amdgcn_wmma_f32_16x16x4_f32(
            false, amr[ks], false, b, (short)0, acc, false, false);
      }
      const int mrow = mbase + ((lane >> 4) << 3);
#pragma unroll
      for (int v = 0; v < 8; ++v) {
        int r = mrow + v;
        if (r < N_EDGE) hR[(size_t)r * DIM + p * 16 + n0] = acc[v];
      }
    }
  }
  __syncthreads();                              // all waves done reading w_real

  // ---- phase 2: w_imag -> h_i ----
  stage_w_issue(wImag, lds);
  stage_w_wait();
  __syncthreads();
  if (active) {
    for (int p = 0; p < 8; ++p) {
      v8f acc = {0.f, 0.f, 0.f, 0.f, 0.f, 0.f, 0.f, 0.f};
      const float* wp = lds + (size_t)koff * DIM + p * 16 + n0;
#pragma unroll
      for (int ks = 0; ks < 32; ++ks) {
        v2f b; b.x = wp[ks * 512]; b.y = wp[ks * 512 + 128];
        acc = __builtin_amdgcn_wmma_f32_16x16x4_f32(
            false, ami[ks], false, b, (short)0, acc, false, false);
      }
      const int mrow = mbase + ((lane >> 4) << 3);
#pragma unroll
      for (int v = 0; v < 8; ++v) {
        int r = mrow + v;
        if (r < N_EDGE) hI[(size_t)r * DIM + p * 16 + n0] = acc[v];
      }
    }
  }
}

// ---------------------------------------------------------------------------
// logit[e] = leaky_relu(h_r[e].a1 + h_i[e].a2, 0.2); atomic segment max on dst.
// One wave per edge (lane holds 4 columns).
// ---------------------------------------------------------------------------
__global__ void logit_kernel(const float* __restrict__ hR, const float* __restrict__ hI,
                             const float* __restrict__ attA, const int* __restrict__ dst,
                             float* __restrict__ logit, float* __restrict__ segmax) {
  const int lane = threadIdx.x & 31;
  const int e = (int)(((size_t)blockIdx.x * blockDim.x + threadIdx.x) >> 5);
  if (e >= N_EDGE) return;
  const float4 a1 = ((const float4*)attA)[lane];
  const float4 a2 = ((const float4*)attA)[32 + lane];
  const float4 r  = ((const float4*)(hR + (size_t)e * DIM))[lane];
  const float4 im = ((const float4*)(hI + (size_t)e * DIM))[lane];
  float s = r.x * a1.x + r.y * a1.y + r.z * a1.z + r.w * a1.w
          + im.x * a2.x + im.y * a2.y + im.z * a2.z + im.w * a2.w;
#pragma unroll
  for (int o = 16; o > 0; o >>= 1) s += __shfl_xor(s, o, 32);
  if (lane == 0) {
    float l = (s > 0.0f) ? s : 0.2f * s;
    logit[e] = l;
    atomic_max_float(&segmax[dst[e]], l);
  }
}

// ---------------------------------------------------------------------------
// ex = exp(logit - segmax[dst]); denom[dst] += ex
// ---------------------------------------------------------------------------
__global__ void softmax_num_kernel(const float* __restrict__ logit,
                                   const int* __restrict__ dst,
                                   const float* __restrict__ segmax,
                                   float* __restrict__ exv, float* __restrict__ denom) {
  int e = blockIdx.x * blockDim.x + threadIdx.x;
  if (e >= N_EDGE) return;
  int d = dst[e];
  float x = __expf(logit[e] - segmax[d]);
  exv[e] = x;
  atomicAdd(&denom[d], x);
}

// ---------------------------------------------------------------------------
// Aggregation: eAcc[dst] += att * h   (one wave per edge, 4 cols/lane)
// ---------------------------------------------------------------------------
__global__ void agg_kernel(const float* __restrict__ hR, const float* __restrict__ hI,
                           const int* __restrict__ dst,
                           const float* __restrict__ exv, const float* __restrict__ denom,
                           float* __restrict__ eAccR, float* __restrict__ eAccI) {
  const int lane = threadIdx.x & 31;
  const int e = (int)(((size_t)blockIdx.x * blockDim.x + threadIdx.x) >> 5);
  if (e >= N_EDGE) return;
  const int d = dst[e];
  const float att = exv[e] / (denom[d] + 1e-10f);
  const float4 r  = ((const float4*)(hR + (size_t)e * DIM))[lane];
  const float4 im = ((const float4*)(hI + (size_t)e * DIM))[lane];
  float* oR = eAccR + (size_t)d * DIM + lane * 4;
  float* oI = eAccI + (size_t)d * DIM + lane * 4;
  atomicAdd(oR + 0, att * r.x);  atomicAdd(oR + 1, att * r.y);
  atomicAdd(oR + 2, att * r.z);  atomicAdd(oR + 3, att * r.w);
  atomicAdd(oI + 0, att * im.x); atomicAdd(oI + 1, att * im.y);
  atomicAdd(oI + 2, att * im.z); atomicAdd(oI + 3, att * im.w);
}

// ---------------------------------------------------------------------------
// In-place tanh over the e_r/e_i accumulator region
// ---------------------------------------------------------------------------
__global__ void tanh_kernel(float* __restrict__ p, size_t n) {
  size_t i = (size_t)blockIdx.x * blockDim.x + threadIdx.x;
  size_t stride = (size_t)gridDim.x * blockDim.x;
  for (size_t j = i; j < n; j += stride) p[j] = tanhf(p[j]);
}

// ---------------------------------------------------------------------------
// Batch gathers: sub_r, sub_i, rel_r, rel_i_q
// ---------------------------------------------------------------------------
__global__ void gather_kernel(const int* __restrict__ sub, const int* __restrict__ rel,
                              const float* __restrict__ eR, const float* __restrict__ eI,
                              const float* __restrict__ rrO, const float* __restrict__ riO,
                              float* __restrict__ out) {
  const int r = blockIdx.x;       // 0..BATCH-1
  const int c = threadIdx.x;      // 0..127
  const int si = sub[r];
  const int ti = rel[r];
  const size_t rc = (size_t)r * DIM + c;
  out[rc]                          = eR[(size_t)si * DIM + c];
  out[(size_t)BATCH * DIM + rc]    = eI[(size_t)si * DIM + c];
  out[2ull * BATCH * DIM + rc]     = rrO[(size_t)ti * DIM + c];
  out[3ull * BATCH * DIM + rc]     = riO[(size_t)ti * DIM + c];
}

// ---------------------------------------------------------------------------
extern "C" void kernel_launch(void* const* d_in, const int* in_sizes, int n_in,
                              void* d_out, int out_size, void* d_ws, size_t ws_size,
                              hipStream_t stream) {
  const int*   edge_index = (const int*)d_in[0];   // [2, N_EDGE]
  const int*   edge_type  = (const int*)d_in[1];
  const int*   sub        = (const int*)d_in[2];
  const int*   rel        = (const int*)d_in[3];
  const float* embR       = (const float*)d_in[4];
  const float* embI       = (const float*)d_in[5];
  const float* relR       = (const float*)d_in[6];
  const float* relI       = (const float*)d_in[7];
  const float* imProj     = (const float*)d_in[8];
  const float* wReal      = (const float*)d_in[9];
  const float* wImag      = (const float*)d_in[10];
  const float* wRel       = (const float*)d_in[11];
  const float* attA       = (const float*)d_in[12];

  const int* edge_src = edge_index;
  const int* edge_dst = edge_index + N_EDGE;

  // ---- output layout (flat, return order) ----
  float* out = (float*)d_out;
  float* eR  = out + 4ull * BATCH * DIM;                 // [N_ENT, DIM]
  float* eI  = eR + (size_t)N_ENT * DIM;                 // [N_ENT, DIM]

  // ---- workspace layout ----
  float* w = (float*)d_ws;
  size_t o = 0;
  float* entI_p = w + o;  o += (size_t)N_ENT * DIM;      // projected ent imag
  float* relI_p = w + o;  o += (size_t)N_REL2 * DIM;     // projected rel imag
  float* hR     = w + o;  o += (size_t)N_EDGE * DIM;
  float* hI     = w + o;  o += (size_t)N_EDGE * DIM;
  float* logit  = w + o;  o += N_EDGE;
  float* exv    = w + o;  o += N_EDGE;
  float* segmax = w + o;  o += N_ENT;
  float* denom  = w + o;  o += N_ENT;
  float* rrOut  = w + o;  o += (size_t)N_REL2 * DIM;
  float* riOut  = w + o;  o += (size_t)N_REL2 * DIM;

  // 1. zero accumulators / softmax state
  init_kernel<<<2048, 256, 0, stream>>>(eR, segmax, denom);

  // 2. imaginary projections through im_proj
  gemm128_kernel<<<(N_ENT + 127) / 128, 256, 0, stream>>>(embI, imProj, entI_p, N_ENT, 0);
  gemm128_kernel<<<(N_REL2 + 127) / 128, 256, 0, stream>>>(relI, imProj, relI_p, N_REL2, 0);

  // 3. per-edge composition + projection GEMMs
  edge_kernel<<<(N_EDGE + 127) / 128, 256, 0, stream>>>(
      edge_src, edge_type, embR, entI_p, relR, relI_p, wReal, wImag, hR, hI);

  // 4. attention logits + segment max (one wave per edge)
  logit_kernel<<<(N_EDGE * 32 + 255) / 256, 256, 0, stream>>>(
      hR, hI, attA, edge_dst, logit, segmax);

  // 5. exp + segment denom
  softmax_num_kernel<<<(N_EDGE + 255) / 256, 256, 0, stream>>>(
      logit, edge_dst, segmax, exv, denom);

  // 6. weighted scatter-add aggregation
  agg_kernel<<<(N_EDGE * 32 + 255) / 256, 256, 0, stream>>>(
      hR, hI, edge_dst, exv, denom, eR, eI);

  // 7. tanh on aggregated entity embeddings (e_r and e_i are contiguous)
  tanh_kernel<<<4096, 256, 0, stream>>>(eR, 2ull * N_ENT * DIM);

  // 8. relation outputs: tanh(rel @ w_rel)
  gemm128_kernel<<<(N_REL2 + 127) / 128, 256, 0, stream>>>(relR, wRel, rrOut, N_REL2, 1);
  gemm128_kernel<<<(N_REL2 + 127) / 128, 256, 0, stream>>>(relI_p, wRel, riOut, N_REL2, 1);

  // 9. batch gathers into the first four output slots
  gather_kernel<<<BATCH, DIM, 0, stream>>>(sub, rel, eR, eI, rrOut, riOut, out);
}